// Net_32787780338386
// MI455X (gfx1250) — compile-verified
//
#include <hip/hip_runtime.h>
#include <hip/hip_bf16.h>
#include <math.h>

#define F_IN 512
#define HID  16
#define NCLS 7

typedef __attribute__((ext_vector_type(2))) float v2f;
typedef __attribute__((ext_vector_type(8))) float v8f;

// ---------------------------------------------------------------------------
// Pack [W1_0 | W1_1 | root1] -> Wcat1 (512 x 48)
// Pack [W2_0(pad8) | W2_1(pad8) | root2(pad8) | 0] -> Wcat2 (16 x 32)
// ---------------------------------------------------------------------------
__global__ void pack_weights(const float* __restrict__ W1, const float* __restrict__ root1,
                             const float* __restrict__ W2, const float* __restrict__ root2,
                             float* __restrict__ Wcat1, float* __restrict__ Wcat2) {
    int i = blockIdx.x * blockDim.x + threadIdx.x;
    if (i < F_IN * 48) {
        int k = i / 48, n = i % 48;
        float v;
        if (n < 16)      v = W1[k * 16 + n];                    // W1[0][k][n]
        else if (n < 32) v = W1[F_IN * 16 + k * 16 + (n - 16)]; // W1[1][k][n-16]
        else             v = root1[k * 16 + (n - 32)];
        Wcat1[i] = v;
    }
    if (i < 16 * 32) {
        int k = i / 32, n = i % 32;
        float v = 0.0f;
        if (n < 7)                  v = W2[k * 7 + n];                 // W2[0]
        else if (n >= 8 && n < 15)  v = W2[16 * 7 + k * 7 + (n - 8)];  // W2[1]
        else if (n >= 16 && n < 23) v = root2[k * 7 + (n - 16)];
        Wcat2[i] = v;
    }
}

__global__ void zero_f32(float* __restrict__ p, long long n) {
    long long i = (long long)blockIdx.x * blockDim.x + threadIdx.x;
    if (i < n) p[i] = 0.0f;
}

// ---------------------------------------------------------------------------
// GEMM1: Hcat[N,48] = X[N,512] @ Wcat1[512,48]   (fp32 WMMA 16x16x4)
// Block: 256 threads = 8 waves; each wave owns 16 rows x 48 cols (3 tiles).
// B is staged in K-paired layout so every B-fragment is one ds_load_b64.
// ---------------------------------------------------------------------------
__global__ __launch_bounds__(256) void gemm1_wmma(const float* __restrict__ X,
                                                  const float* __restrict__ W,
                                                  float* __restrict__ Hcat, int nrows) {
    __shared__ __align__(16) float As[128 * 32];  // 16 KB
    __shared__ __align__(16) float Bs[32 * 48];   // 6 KB, paired: [(k/2)][n][k&1]
    const int tid  = threadIdx.x;
    const int wave = tid >> 5;
    const int lane = tid & 31;
    const int half = lane >> 4;   // 0: lanes 0-15, 1: lanes 16-31
    const int l16  = lane & 15;
    const int row_base = blockIdx.x * 128;

    v8f acc0 = {}, acc1 = {}, acc2 = {};

    for (int kb = 0; kb < F_IN; kb += 32) {
        // Stage A tile (128 rows x 32 K) with coalesced 128-bit loads
        #pragma unroll
        for (int it = 0; it < 4; ++it) {
            int idx = tid + it * 256;        // 1024 float4 slots
            int r   = idx >> 3;              // row 0..127
            int kq  = (idx & 7) << 2;        // 0,4,...,28
            int gr  = row_base + r; if (gr >= nrows) gr = nrows - 1;
            float4 v = *(const float4*)(X + (size_t)gr * F_IN + kb + kq);
            *(float4*)(As + r * 32 + kq) = v;
        }
        // Stage B tile (32 K x 48 cols) into K-paired layout:
        // element (k, n) -> Bs[((k>>1)*48 + n)*2 + (k&1)]
        for (int i = tid; i < 32 * 48; i += 256) {
            int k = i / 48, n = i % 48;
            Bs[(((unsigned)k >> 1) * 48 + n) * 2 + (k & 1)] = W[(kb + k) * 48 + n];
        }
        __syncthreads();

        const int arow = wave * 16 + l16;    // A-frag: M = l16 in both lane halves
        #pragma unroll
        for (int kk = 0; kk < 32; kk += 4) {
            const int kq = kk + half * 2;    // this lane-half's K pair: (kq, kq+1)
            v2f a  = *(const v2f*)(As + arow * 32 + kq);
            const float* bp = Bs + ((unsigned)kq >> 1) * 96;   // *48*2
            v2f b0 = *(const v2f*)(bp + (l16      ) * 2);
            v2f b1 = *(const v2f*)(bp + (l16 + 16 ) * 2);
            v2f b2 = *(const v2f*)(bp + (l16 + 32 ) * 2);
            acc0 = __builtin_amdgcn_wmma_f32_16x16x4_f32(false, a, false, b0, (short)0, acc0, false, false);
            acc1 = __builtin_amdgcn_wmma_f32_16x16x4_f32(false, a, false, b1, (short)0, acc1, false, false);
            acc2 = __builtin_amdgcn_wmma_f32_16x16x4_f32(false, a, false, b2, (short)0, acc2, false, false);
        }
        __syncthreads();
    }
    // D layout: VGPR r -> lanes 0-15 M=r, lanes 16-31 M=8+r
    #pragma unroll
    for (int r = 0; r < 8; ++r) {
        int grow = row_base + wave * 16 + half * 8 + r;
        if (grow < nrows) {
            float* o = Hcat + (size_t)grow * 48 + l16;
            o[0]  = acc0[r];
            o[16] = acc1[r];
            o[32] = acc2[r];
        }
    }
}

// ---------------------------------------------------------------------------
// GEMM2: Hcat2[N,32] = h[N,16] @ Wcat2[16,32]   (fp32 WMMA 16x16x4, K=16)
// ---------------------------------------------------------------------------
__global__ __launch_bounds__(256) void gemm2_wmma(const float* __restrict__ Hin,
                                                  const float* __restrict__ W,
                                                  float* __restrict__ Hcat2, int nrows) {
    __shared__ __align__(16) float As[128 * 16]; // 8 KB
    __shared__ __align__(16) float Bs[16 * 32];  // 2 KB, paired layout
    const int tid  = threadIdx.x;
    const int wave = tid >> 5;
    const int lane = tid & 31;
    const int half = lane >> 4;
    const int l16  = lane & 15;
    const int row_base = blockIdx.x * 128;

    #pragma unroll
    for (int it = 0; it < 2; ++it) {
        int idx = tid + it * 256;       // 512 float4 slots
        int r   = idx >> 2;             // row 0..127
        int kq  = (idx & 3) << 2;
        int gr  = row_base + r; if (gr >= nrows) gr = nrows - 1;
        float4 v = *(const float4*)(Hin + (size_t)gr * 16 + kq);
        *(float4*)(As + r * 16 + kq) = v;
    }
    for (int i = tid; i < 16 * 32; i += 256) {
        int k = i / 32, n = i % 32;
        Bs[(((unsigned)k >> 1) * 32 + n) * 2 + (k & 1)] = W[i];
    }
    __syncthreads();

    v8f acc0 = {}, acc1 = {};
    const int arow = wave * 16 + l16;
    #pragma unroll
    for (int kk = 0; kk < 16; kk += 4) {
        const int kq = kk + half * 2;
        v2f a  = *(const v2f*)(As + arow * 16 + kq);
        const float* bp = Bs + ((unsigned)kq >> 1) * 64;   // *32*2
        v2f b0 = *(const v2f*)(bp + (l16     ) * 2);
        v2f b1 = *(const v2f*)(bp + (l16 + 16) * 2);
        acc0 = __builtin_amdgcn_wmma_f32_16x16x4_f32(false, a, false, b0, (short)0, acc0, false, false);
        acc1 = __builtin_amdgcn_wmma_f32_16x16x4_f32(false, a, false, b1, (short)0, acc1, false, false);
    }
    #pragma unroll
    for (int r = 0; r < 8; ++r) {
        int grow = row_base + wave * 16 + half * 8 + r;
        if (grow < nrows) {
            float* o = Hcat2 + (size_t)grow * 32 + l16;
            o[0]  = acc0[r];
            o[16] = acc1[r];
        }
    }
}

// ---------------------------------------------------------------------------
// Edge scatter, layer 1: 16 lanes per edge (one per feature).
// msg = h0[src]*b0 + h1[src]*b1 ; atomic scatter to agg[dst]; lane 0 bumps deg.
// ---------------------------------------------------------------------------
__global__ __launch_bounds__(256) void edge1_kernel(const float* __restrict__ Hcat,
                                                    const float* __restrict__ u,
                                                    const int* __restrict__ src,
                                                    const int* __restrict__ dst,
                                                    float* __restrict__ agg,
                                                    float* __restrict__ deg,
                                                    long long E) {
    long long t = (long long)blockIdx.x * 256 + threadIdx.x;
    long long e = t >> 4;
    int f = (int)(t & 15);
    if (e >= E) return;
    float b1 = u[e];
    b1 = fminf(fmaxf(b1, 0.0f), 1.0f);
    float b0 = 1.0f - b1;
    int s = src[e], d = dst[e];
    const float* hs = Hcat + (size_t)s * 48;
    float m = hs[f] * b0 + hs[16 + f] * b1;
    atomicAdd(agg + (size_t)d * 16 + f, m);
    if (f == 0) atomicAdd(deg + d, 1.0f);
}

// ---------------------------------------------------------------------------
// Node update, layer 1: mean-agg + root + bias -> ELU -> h[N,16]
// ---------------------------------------------------------------------------
__global__ __launch_bounds__(256) void node1_kernel(const float* __restrict__ agg,
                                                    const float* __restrict__ deg,
                                                    const float* __restrict__ Hcat,
                                                    const float* __restrict__ bias1,
                                                    float* __restrict__ h, int n_nodes) {
    int idx = blockIdx.x * 256 + threadIdx.x;
    int n = idx >> 4, f = idx & 15;
    if (n >= n_nodes) return;
    float dg = fmaxf(deg[n], 1.0f);
    float a = agg[(size_t)n * 16 + f] / dg + Hcat[(size_t)n * 48 + 32 + f] + bias1[f];
    h[(size_t)n * 16 + f] = (a > 0.0f) ? a : expm1f(a);
}

// ---------------------------------------------------------------------------
// Edge scatter, layer 2: 8 lanes per edge (7 classes + 1 idle). Deg reused.
// ---------------------------------------------------------------------------
__global__ __launch_bounds__(256) void edge2_kernel(const float* __restrict__ Hcat2,
                                                    const float* __restrict__ u,
                                                    const int* __restrict__ src,
                                                    const int* __restrict__ dst,
                                                    float* __restrict__ agg2,
                                                    long long E) {
    long long t = (long long)blockIdx.x * 256 + threadIdx.x;
    long long e = t >> 3;
    int f = (int)(t & 7);
    if (e >= E || f >= NCLS) return;
    float b1 = u[e];
    b1 = fminf(fmaxf(b1, 0.0f), 1.0f);
    float b0 = 1.0f - b1;
    int s = src[e], d = dst[e];
    const float* hs = Hcat2 + (size_t)s * 32;
    float m = hs[f] * b0 + hs[8 + f] * b1;
    atomicAdd(agg2 + (size_t)d * 8 + f, m);
}

// ---------------------------------------------------------------------------
// Final: mean-agg + root + bias -> log_softmax over 7 classes.
// ---------------------------------------------------------------------------
__global__ __launch_bounds__(256) void final_kernel(const float* __restrict__ agg2,
                                                    const float* __restrict__ deg,
                                                    const float* __restrict__ Hcat2,
                                                    const float* __restrict__ bias2,
                                                    float* __restrict__ out, int n_nodes) {
    int n = blockIdx.x * 256 + threadIdx.x;
    if (n >= n_nodes) return;
    float dg = fmaxf(deg[n], 1.0f);
    float v[NCLS];
    float mx = -1e30f;
    #pragma unroll
    for (int c = 0; c < NCLS; ++c) {
        float t = agg2[(size_t)n * 8 + c] / dg + Hcat2[(size_t)n * 32 + 16 + c] + bias2[c];
        v[c] = t;
        mx = fmaxf(mx, t);
    }
    float s = 0.0f;
    #pragma unroll
    for (int c = 0; c < NCLS; ++c) s += __expf(v[c] - mx);
    float lse = logf(s);
    #pragma unroll
    for (int c = 0; c < NCLS; ++c) out[(size_t)n * NCLS + c] = v[c] - mx - lse;
}

// ---------------------------------------------------------------------------
extern "C" void kernel_launch(void* const* d_in, const int* in_sizes, int n_in,
                              void* d_out, int out_size, void* d_ws, size_t ws_size,
                              hipStream_t stream) {
    const float* x     = (const float*)d_in[0];
    const float* eattr = (const float*)d_in[1];
    const int*   src   = (const int*)d_in[2];
    const int*   dst   = (const int*)d_in[3];
    const float* W1    = (const float*)d_in[4];
    const float* root1 = (const float*)d_in[5];
    const float* bias1 = (const float*)d_in[6];
    const float* W2    = (const float*)d_in[7];
    const float* root2 = (const float*)d_in[8];
    const float* bias2 = (const float*)d_in[9];
    float* out = (float*)d_out;

    const int       N = in_sizes[0] / F_IN;
    const long long E = (long long)in_sizes[2];

    // Workspace layout (floats). agg/deg/agg2 are contiguous (one zero pass).
    float* ws    = (float*)d_ws;
    float* Wcat1 = ws;                          // 512*48
    float* Wcat2 = Wcat1 + F_IN * 48;           // 16*32
    float* Hcat  = Wcat2 + 16 * 32;             // N*48
    float* h     = Hcat  + (size_t)N * 48;      // N*16
    float* Hcat2 = h     + (size_t)N * 16;      // N*32
    float* agg   = Hcat2 + (size_t)N * 32;      // N*16 (zero region start)
    float* deg   = agg   + (size_t)N * 16;      // N
    float* agg2  = deg   + (size_t)N;           // N*8  (zero region end)

    // 0) pack weights, zero accumulators (d_ws is poisoned by harness)
    pack_weights<<<(F_IN * 48 + 255) / 256, 256, 0, stream>>>(W1, root1, W2, root2, Wcat1, Wcat2);
    long long nz = (long long)N * 25;
    zero_f32<<<(unsigned)((nz + 255) / 256), 256, 0, stream>>>(agg, nz);

    // 1) layer-1 fused GEMM (WMMA f32): Hcat = X @ [W1_0|W1_1|root1]
    gemm1_wmma<<<(N + 127) / 128, 256, 0, stream>>>(x, Wcat1, Hcat, N);

    // 2) edge gather/scatter (Hcat is L2-resident: 38.4 MB < 192 MB L2)
    long long t1 = E * 16;
    edge1_kernel<<<(unsigned)((t1 + 255) / 256), 256, 0, stream>>>(Hcat, eattr, src, dst, agg, deg, E);

    // 3) node update + ELU
    node1_kernel<<<(N * 16 + 255) / 256, 256, 0, stream>>>(agg, deg, Hcat, bias1, h, N);

    // 4) layer-2 fused GEMM (WMMA f32): Hcat2 = h @ [W2_0|W2_1|root2] (padded)
    gemm2_wmma<<<(N + 127) / 128, 256, 0, stream>>>(h, Wcat2, Hcat2, N);

    // 5) edge gather/scatter layer 2 (deg reused)
    long long t2 = E * 8;
    edge2_kernel<<<(unsigned)((t2 + 255) / 256), 256, 0, stream>>>(Hcat2, eattr, src, dst, agg2, E);

    // 6) final node update + log_softmax
    final_kernel<<<(N + 255) / 256, 256, 0, stream>>>(agg2, deg, Hcat2, bias2, out, N);
}